// Resizer_1726576856659
// MI455X (gfx1250) — compile-verified
//
#include <hip/hip_runtime.h>

typedef float v2f __attribute__((ext_vector_type(2)));
typedef float v8f __attribute__((ext_vector_type(8)));

#define NC   48      // N*C
#define H    1024
#define W    1024
#define OH   512
#define OW   512
#define P    8       // taps per output
#define GRP  32      // groups of 16 outputs per resized dim (512/16)
#define WIN  40      // dense K window per group (multiple of 4, covers 38-tap span)
#define KCH  (WIN/4) // 10 wmma K-chunks of 4
#define TROW 1028    // LDS row stride (dwords): 1028%64==4 -> conflict-free

// Window base for output group g (outputs 16g..16g+15): taps span 32g-3 .. 32g+34.
__device__ __forceinline__ int group_base(int g) {
    int b = 32 * g - 4;
    if (b < 0) b = 0;
    if (b > H - WIN) b = H - WIN;
    return b;
}

// Densify the 8-tap sparse weights into per-group 16x40 (A-side) and 40x16
// (B-side, pre-transposed) tiles. += absorbs mirror-duplicated border taps.
__global__ void build_weights(const float* __restrict__ w0,
                              const long long* __restrict__ fov0,
                              const float* __restrict__ w1,
                              const long long* __restrict__ fov1,
                              float* __restrict__ Wv,   // [GRP][16][WIN]
                              float* __restrict__ Wh)   // [GRP][WIN][16]
{
    int t = blockIdx.x * blockDim.x + threadIdx.x;
    if (t >= OH) return;
    int g = t >> 4, i = t & 15;
    int base = group_base(g);
    float* wv = Wv + (size_t)(g * 16 + i) * WIN;
    float* wh = Wh + (size_t)g * WIN * 16 + i;
    for (int k = 0; k < WIN; ++k) wv[k] = 0.0f;
    for (int k = 0; k < WIN; ++k) wh[k * 16] = 0.0f;
    for (int p = 0; p < P; ++p) {
        int jv = (int)fov0[p * OH + t] - base;
        if (jv >= 0 && jv < WIN) wv[jv] += w0[p * OH + t];
        int jh = (int)fov1[p * OH + t] - base;
        if (jh >= 0 && jh < WIN) wh[jh * 16] += w1[p * OH + t];
    }
}

// Fused separable resize. One workgroup (8 waves) per (nc, vertical group g):
//  Phase 1: 40-row global window -> 16x1024 vertically-resized strip in LDS
//           (one WMMA tile chain per 16-column slab, 8 slabs per wave).
//  Phase 2: horizontal resize of the LDS strip -> 16x512 output rows
//           (A operand from LDS via ds_load, B = pre-transposed weights).
__global__ void __launch_bounds__(256)
resize_fused(const float* __restrict__ X,   // [NC][H][W]
             const float* __restrict__ Wv,  // [GRP][16][WIN]
             const float* __restrict__ Wh,  // [GRP][WIN][16]
             float* __restrict__ O)         // [NC][OH][OW]
{
    __shared__ float tmp[16 * TROW];

    int wid  = threadIdx.x >> 5;
    int lane = threadIdx.x & 31;
    int g  = blockIdx.x & (GRP - 1);
    int nc = blockIdx.x / GRP;
    int baseV = group_base(g);
    int m  = lane & 15;
    int hi = lane >> 4;           // lane half selects K parity pair

    // ---- Phase 1: contract over H ----
    const float* wrow = Wv + (size_t)(g * 16 + m) * WIN + 2 * hi;
    v2f aw[KCH];
#pragma unroll
    for (int k = 0; k < KCH; ++k) {
        aw[k].x = wrow[4 * k + 0];
        aw[k].y = wrow[4 * k + 1];
    }
    const float* xbase = X + (size_t)nc * H * W + (size_t)(baseV + 2 * hi) * W + m;
#pragma unroll 2
    for (int t = 0; t < 8; ++t) {
        int xt = wid + 8 * t;                 // 64 column slabs / 8 waves
        const float* xcol = xbase + xt * 16;
        v8f c = {0.f, 0.f, 0.f, 0.f, 0.f, 0.f, 0.f, 0.f};
#pragma unroll
        for (int k = 0; k < KCH; ++k) {
            v2f b;
            b.x = xcol[(size_t)(4 * k + 0) * W];
            b.y = xcol[(size_t)(4 * k + 1) * W];
            c = __builtin_amdgcn_wmma_f32_16x16x4_f32(false, aw[k], false, b,
                                                      (short)0, c, false, false);
        }
#pragma unroll
        for (int r = 0; r < 8; ++r)
            tmp[(r + 8 * hi) * TROW + xt * 16 + m] = c[r];  // banks 4r+c / 4r+32+c
    }

    __syncthreads();

    // ---- Phase 2: contract over W (transposed read served by LDS) ----
#pragma unroll 2
    for (int t = 0; t < 4; ++t) {
        int og = wid + 8 * t;                 // 32 output groups / 8 waves
        int baseH = group_base(og);
        const float* arow = &tmp[m * TROW + baseH + 2 * hi];
        const float* bcol = Wh + (size_t)og * WIN * 16 + (2 * hi) * 16 + m;
        v8f c = {0.f, 0.f, 0.f, 0.f, 0.f, 0.f, 0.f, 0.f};
#pragma unroll
        for (int k = 0; k < KCH; ++k) {
            v2f a, b;
            a.x = arow[4 * k + 0];            // merges to ds_load_b64, 8B aligned
            a.y = arow[4 * k + 1];            // all 64 LDS banks hit exactly once
            b.x = bcol[(4 * k + 0) * 16];
            b.y = bcol[(4 * k + 1) * 16];
            c = __builtin_amdgcn_wmma_f32_16x16x4_f32(false, a, false, b,
                                                      (short)0, c, false, false);
        }
        float* out = O + (size_t)nc * OH * OW + (size_t)(g * 16 + 8 * hi) * OW
                       + og * 16 + m;
#pragma unroll
        for (int r = 0; r < 8; ++r) out[(size_t)r * OW] = c[r];
    }
}

extern "C" void kernel_launch(void* const* d_in, const int* in_sizes, int n_in,
                              void* d_out, int out_size, void* d_ws, size_t ws_size,
                              hipStream_t stream) {
    const float*     x    = (const float*)d_in[0];      // (16,3,1024,1024) f32
    const float*     w0   = (const float*)d_in[1];      // (8,512,1,1,1)    f32
    const float*     w1   = (const float*)d_in[2];      // (8,512,1,1,1)    f32
    const long long* fov0 = (const long long*)d_in[3];  // (8,512)          i64
    const long long* fov1 = (const long long*)d_in[4];  // (8,512)          i64
    float* out = (float*)d_out;                         // (16,3,512,512)   f32

    float* Wv = (float*)d_ws;                 // 80 KB
    float* Wh = Wv + GRP * 16 * WIN;          // 80 KB

    build_weights<<<1, 512, 0, stream>>>(w0, fov0, w1, fov1, Wv, Wh);
    resize_fused<<<NC * GRP, 256, 0, stream>>>(x, Wv, Wh, out);
}